// Capsule_10668698763531
// MI455X (gfx1250) — compile-verified
//
#include <hip/hip_runtime.h>
#include <hip/hip_bf16.h>

typedef __attribute__((ext_vector_type(2))) float v2f;
typedef __attribute__((ext_vector_type(8))) float v8f;
typedef int v4i __attribute__((vector_size(4 * sizeof(int))));

#define NC    16
#define DC    32
#define SEQ   512
#define DIN   256
#define NOUT  512          // NC*DC
#define MTOT  131072       // B*SEQ = 256*512

#define BLK_M 64
#define BLK_N 128
#define BLK_K 32
#define NKB   (DIN / BLK_K)

// ---------------------------------------------------------------------------
// Async global->LDS copy (CDNA5 GLOBAL_LOAD_ASYNC_TO_LDS_B128, ASYNCcnt),
// with a synchronous fallback if the builtin is unavailable.
// ---------------------------------------------------------------------------
#if defined(__has_builtin)
#  if __has_builtin(__builtin_amdgcn_global_load_async_to_lds_b128)
#    define HAVE_ASYNC_LDS 1
#  endif
#endif
#ifndef HAVE_ASYNC_LDS
#  define HAVE_ASYNC_LDS 0
#endif

__device__ __forceinline__ void copy16_to_lds(const float* g, float* l) {
#if HAVE_ASYNC_LDS
    __builtin_amdgcn_global_load_async_to_lds_b128(
        (__attribute__((address_space(1))) v4i*)g,
        (__attribute__((address_space(3))) v4i*)l,
        /*offset=*/0, /*cpol=*/0);
#else
    *(float4*)l = *(const float4*)g;
#endif
}

__device__ __forceinline__ void async_copy_wait() {
#if HAVE_ASYNC_LDS
#  if __has_builtin(__builtin_amdgcn_s_wait_asynccnt)
    __builtin_amdgcn_s_wait_asynccnt(0);
#  else
    asm volatile("s_wait_asynccnt 0x0" ::: "memory");
#  endif
#endif
}

// ---------------------------------------------------------------------------
// Kernel 1: u_hat = u_vecs @ W   (M=131072, K=256, N=512), fp32 WMMA 16x16x4,
// double-buffered LDS with async global->LDS staging.
// Output row-major: Uhat[(b*512+i)*512 + (n*32+d)]
// ---------------------------------------------------------------------------
__global__ __launch_bounds__(256) void capsule_gemm(const float* __restrict__ A,
                                                    const float* __restrict__ W,
                                                    float* __restrict__ Uhat) {
    __shared__ float As[2][BLK_M][BLK_K + 1];  // pad: conflict-free column reads
    __shared__ float Bs[2][BLK_K][BLK_N];

    const int t    = threadIdx.x;
    const int wid  = t >> 5;
    const int lane = t & 31;
    const int half = lane >> 4;      // 0: lanes 0-15, 1: lanes 16-31
    const int lq   = lane & 15;
    const int wm   = wid >> 2;       // 2 x 4 wave grid, wave tile 32x32
    const int wn   = wid & 3;

    const long rowBase = (long)blockIdx.x * BLK_M;
    const int  colBase = blockIdx.y * BLK_N;

    v8f acc[2][2] = {};

    // Stage one K-block (6 x 16B per thread: 2 for A-tile, 4 for B-tile)
    auto stage = [&](int buf, int kb) {
        #pragma unroll
        for (int it = 0; it < 2; ++it) {
            int f  = it * 256 + t;
            int r  = f >> 3;             // 8 float4 per A row (32 floats)
            int c4 = (f & 7) << 2;
            copy16_to_lds(A + (rowBase + r) * DIN + kb + c4, &As[buf][r][c4]);
        }
        #pragma unroll
        for (int it = 0; it < 4; ++it) {
            int f  = it * 256 + t;
            int r  = f >> 5;             // 32 float4 per B row (128 floats)
            int c4 = (f & 31) << 2;
            copy16_to_lds(W + (long)(kb + r) * NOUT + colBase + c4, &Bs[buf][r][c4]);
        }
    };

    stage(0, 0);

    for (int kb = 0; kb < NKB; ++kb) {
        const int cur = kb & 1;
        async_copy_wait();               // this wave's loads for 'cur' are in LDS
        __syncthreads();                 // ...and everyone else's too
        if (kb + 1 < NKB) stage(cur ^ 1, (kb + 1) * BLK_K);

        #pragma unroll
        for (int kk = 0; kk < BLK_K; kk += 4) {
            v2f af[2], bf[2];
            #pragma unroll
            for (int mt = 0; mt < 2; ++mt) {
                int m = wm * 32 + mt * 16 + lq;      // A: lanes 0-15 K0/K1, 16-31 K2/K3
                af[mt].x = As[cur][m][kk + half * 2];
                af[mt].y = As[cur][m][kk + half * 2 + 1];
            }
            #pragma unroll
            for (int nt = 0; nt < 2; ++nt) {
                int c = wn * 32 + nt * 16 + lq;      // B: K rows striped across lanes
                bf[nt].x = Bs[cur][kk + half * 2][c];
                bf[nt].y = Bs[cur][kk + half * 2 + 1][c];
            }
            #pragma unroll
            for (int mt = 0; mt < 2; ++mt)
                #pragma unroll
                for (int nt = 0; nt < 2; ++nt)
                    acc[mt][nt] = __builtin_amdgcn_wmma_f32_16x16x4_f32(
                        false, af[mt], false, bf[nt], (short)0, acc[mt][nt], false, false);
        }
        __syncthreads();                 // all reads of 'cur' done before overwrite
    }

    // Epilogue: C/D layout -> VGPR v holds rows M=v (lanes 0-15), M=v+8 (lanes 16-31)
    #pragma unroll
    for (int mt = 0; mt < 2; ++mt)
        #pragma unroll
        for (int nt = 0; nt < 2; ++nt) {
            long r0 = rowBase + wm * 32 + mt * 16 + half * 8;
            int  c  = colBase + wn * 32 + nt * 16 + lq;
            #pragma unroll
            for (int v = 0; v < 8; ++v)
                Uhat[(r0 + v) * (long)NOUT + c] = acc[mt][nt][v];
        }
}

// ---------------------------------------------------------------------------
// Kernel 2: fused dynamic routing (3 rounds -> 3 passes over u_hat[b]).
// One workgroup (8 waves) per batch element.
// ---------------------------------------------------------------------------
__global__ __launch_bounds__(256) void capsule_route(const float* __restrict__ Uhat,
                                                     float* __restrict__ Out) {
    const int b = blockIdx.x;
    const float* U = Uhat + (long)b * SEQ * NOUT;

    __shared__ float prev[NOUT];       // normalized outputs of previous round
    __shared__ float red[NOUT];        // reduction buffer
    __shared__ float accs[8][NOUT];    // per-wave partial accumulators
    __shared__ float scale[NC];

    const int t    = threadIdx.x;
    const int wid  = t >> 5;
    const int lane = t & 31;
    const int c0   = 2 * t;            // this thread's 2 columns for reductions

    // ---- Round 0: uniform softmax => outputs0 = normalize(sum_i u_hat) ----
    float s0 = 0.f, s1 = 0.f;
    for (int i = 0; i < SEQ; ++i) {
        float2 v = *(const float2*)(U + (long)i * NOUT + c0);
        s0 += v.x; s1 += v.y;
    }
    red[c0] = s0; red[c0 + 1] = s1;
    __syncthreads();
    if (t < NC) {
        float ss = 0.f;
        #pragma unroll
        for (int d = 0; d < DC; ++d) { float x = red[t * DC + d]; ss += x * x; }
        scale[t] = rsqrtf(ss + 1e-7f);
    }
    __syncthreads();
    prev[c0]     = red[c0]     * scale[c0 >> 5];
    prev[c0 + 1] = red[c0 + 1] * scale[(c0 + 1) >> 5];
    __syncthreads();

    // ---- Rounds 1,2: fused b-update + softmax(over n) + weighted sum ----
    for (int iter = 0; iter < 2; ++iter) {
        float acc[16];
        #pragma unroll
        for (int j = 0; j < 16; ++j) acc[j] = 0.f;
        const int cb = lane * 16;          // lane owns 16 cols = half of capsule n=lane>>1

        for (int i = wid; i < SEQ; i += 8) {
            const float* rp = U + (long)i * NOUT + cb;
            float r[16];
            #pragma unroll
            for (int q = 0; q < 4; ++q) {
                float4 v = *(const float4*)(rp + q * 4);
                r[q * 4] = v.x; r[q * 4 + 1] = v.y; r[q * 4 + 2] = v.z; r[q * 4 + 3] = v.w;
            }
            // logit b[n,i] = <outputs_prev[n], u_hat[n,i,:]> (pairwise lanes share n)
            float part = 0.f;
            #pragma unroll
            for (int j = 0; j < 16; ++j) part += r[j] * prev[cb + j];
            part += __shfl_xor(part, 1, 32);
            // softmax over the 16 capsules (each value duplicated across lane pairs)
            float mx = part;
            #pragma unroll
            for (int off = 16; off >= 1; off >>= 1) mx = fmaxf(mx, __shfl_xor(mx, off, 32));
            float e  = __expf(part - mx);
            float sm = e;
            #pragma unroll
            for (int off = 16; off >= 1; off >>= 1) sm += __shfl_xor(sm, off, 32);
            float cw = e * (2.0f / sm);    // wave sum counts each capsule twice
            #pragma unroll
            for (int j = 0; j < 16; ++j) acc[j] += cw * r[j];
        }
        #pragma unroll
        for (int j = 0; j < 16; ++j) accs[wid][cb + j] = acc[j];
        __syncthreads();

        float a0 = 0.f, a1 = 0.f;
        #pragma unroll
        for (int w = 0; w < 8; ++w) { a0 += accs[w][c0]; a1 += accs[w][c0 + 1]; }
        red[c0] = a0; red[c0 + 1] = a1;
        __syncthreads();
        if (t < NC) {
            float ss = 0.f;
            #pragma unroll
            for (int d = 0; d < DC; ++d) { float x = red[t * DC + d]; ss += x * x; }
            scale[t] = rsqrtf(ss + 1e-7f);
        }
        __syncthreads();
        prev[c0]     = red[c0]     * scale[c0 >> 5];
        prev[c0 + 1] = red[c0 + 1] * scale[(c0 + 1) >> 5];
        __syncthreads();
    }

    // Output: (B, 16, 32) flat; column index == n*32+d already
    float* ob = Out + (long)b * NOUT;
    ob[c0]     = prev[c0];
    ob[c0 + 1] = prev[c0 + 1];
}

// ---------------------------------------------------------------------------
extern "C" void kernel_launch(void* const* d_in, const int* in_sizes, int n_in,
                              void* d_out, int out_size, void* d_ws, size_t ws_size,
                              hipStream_t stream) {
    const float* u_vecs = (const float*)d_in[0];   // (256, 512, 256) f32
    const float* W      = (const float*)d_in[1];   // (1, 256, 512) f32
    float*       out    = (float*)d_out;           // (256, 16, 32) f32
    float*       uhat   = (float*)d_ws;            // needs 256*512*512*4 = 256 MB

    dim3 g1(MTOT / BLK_M, NOUT / BLK_N);           // 2048 x 4
    capsule_gemm<<<g1, 256, 0, stream>>>(u_vecs, W, uhat);
    capsule_route<<<256, 256, 0, stream>>>(uhat, out);
}